// GraphEncoder_44341242364692
// MI455X (gfx1250) — compile-verified
//
#include <hip/hip_runtime.h>
#include <hip/hip_bf16.h>

typedef __attribute__((ext_vector_type(2))) float v2f;
typedef __attribute__((ext_vector_type(8))) float v8f;

#define N_NODES 100000
#define N_EDGES 1600000
#define N_GRAPHS 64

// ---------------------------------------------------------------------------
// Edge scatter-add: agg[dst[e]][c] += x[src[e]][c]; d = 1<<shift (128 or 256)
// ---------------------------------------------------------------------------
__global__ __launch_bounds__(256)
void scatter_add_kernel(const float* __restrict__ x,
                        const int* __restrict__ src,
                        const int* __restrict__ dst,
                        float* __restrict__ agg,
                        int nE, int shift) {
    long long idx = (long long)blockIdx.x * blockDim.x + threadIdx.x;
    int d = 1 << shift;
    int e = (int)(idx >> shift);
    if (e >= nE) return;
    int c = (int)(idx & (d - 1));
    atomicAdd(&agg[(size_t)dst[e] * d + c], x[(size_t)src[e] * d + c]);
}

// ---------------------------------------------------------------------------
// Fused: C[16-row strip, DOUT] = LNReLU( A@Wrel^T + X@Wroot^T + bias )
// Block = 128 threads = 4 waves; each wave owns DOUT/4 columns as NT=DOUT/64
// 16x16 WMMA tiles (V_WMMA_F32_16X16X4_F32, exact f32).
// A-fragment (16x4 f32): lanes 0-15: M=lane, v0=K0 v1=K1; lanes 16-31:
// M=lane-16, v0=K2 v1=K3. B (4x16) mirrors row-striped across lanes.
// C/D: vgpr i -> M = i + (lane>=16 ? 8 : 0), N = lane&15.
// Epilogue: accumulators (+bias) -> LDS tile, two-level LDS reduction for
// mean/var per row, then normalize+scale+shift+ReLU, coalesced global store.
// ---------------------------------------------------------------------------
template <int DOUT>
__global__ __launch_bounds__(128)
void gemm_ln_relu_kernel(const float* __restrict__ A,     // [N,K] agg
                         const float* __restrict__ X,     // [N,K] root input
                         const float* __restrict__ Wrel,  // [DOUT,K]
                         const float* __restrict__ Wroot, // [DOUT,K]
                         const float* __restrict__ bias,  // [DOUT]
                         const float* __restrict__ g,     // [DOUT]
                         const float* __restrict__ be,    // [DOUT]
                         float* __restrict__ C,           // [N,DOUT]
                         int K) {
    constexpr int NT = DOUT / 64;            // 16-col tiles per wave (2 or 4)
    const int tileM   = blockIdx.x;
    const int tid     = threadIdx.x;
    const int wave    = tid >> 5;
    const int lane    = tid & 31;
    const int m       = lane & 15;           // M (and N) index within tile
    const int kh      = (lane >> 4) << 1;    // 0 or 2: K sub-offset this lane
    const int mbase   = (lane >> 4) * 8;
    const int colbase = wave * (DOUT / 4);

    __shared__ float tile[16 * DOUT];
    __shared__ float psum[16 * 8];
    __shared__ float psq[16 * 8];
    __shared__ float mu_s[16];
    __shared__ float rs_s[16];

    const size_t rowA = (size_t)(tileM * 16 + m) * K;

    v8f acc[NT];
    #pragma unroll
    for (int j = 0; j < NT; ++j) acc[j] = (v8f){};

    #pragma unroll 1
    for (int pass = 0; pass < 2; ++pass) {
        const float* __restrict__ Mx = pass ? X : A;
        const float* __restrict__ Wx = pass ? Wroot : Wrel;
        #pragma unroll 2
        for (int k0 = 0; k0 < K; k0 += 4) {
            v2f a;
            a.x = Mx[rowA + k0 + kh];
            a.y = Mx[rowA + k0 + kh + 1];
            #pragma unroll
            for (int j = 0; j < NT; ++j) {
                const size_t rowW = (size_t)(colbase + j * 16 + m) * K;
                v2f b;
                b.x = Wx[rowW + k0 + kh];
                b.y = Wx[rowW + k0 + kh + 1];
                acc[j] = __builtin_amdgcn_wmma_f32_16x16x4_f32(
                    /*neg_a=*/false, a, /*neg_b=*/false, b,
                    /*c_mod=*/(short)0, acc[j], /*reuse_a=*/false, /*reuse_b=*/false);
            }
        }
    }

    // Accumulators + bias -> LDS tile (bias must precede LN statistics)
    #pragma unroll
    for (int j = 0; j < NT; ++j) {
        const int col = colbase + j * 16 + m;
        const float bv = bias[col];
        #pragma unroll
        for (int i = 0; i < 8; ++i)
            tile[(mbase + i) * DOUT + col] = acc[j][i] + bv;
    }
    __syncthreads();

    // Per-row mean / variance: 8 partial threads per row, then 16 finalizers
    {
        const int r = tid >> 3;              // 0..15
        const int p = tid & 7;               // 0..7
        float s = 0.0f, q = 0.0f;
        for (int c = p; c < DOUT; c += 8) {
            const float v = tile[r * DOUT + c];
            s += v;
            q += v * v;
        }
        psum[r * 8 + p] = s;
        psq[r * 8 + p]  = q;
    }
    __syncthreads();
    if (tid < 16) {
        float s = 0.0f, q = 0.0f;
        #pragma unroll
        for (int p = 0; p < 8; ++p) {
            s += psum[tid * 8 + p];
            q += psq[tid * 8 + p];
        }
        const float mu  = s * (1.0f / DOUT);
        const float var = q * (1.0f / DOUT) - mu * mu;
        mu_s[tid] = mu;
        rs_s[tid] = rsqrtf(var + 1e-5f);
    }
    __syncthreads();

    // Normalize + affine + ReLU, coalesced store
    for (int idx = tid; idx < 16 * DOUT; idx += 128) {
        const int r = idx / DOUT;
        const int c = idx - r * DOUT;
        float y = (tile[idx] - mu_s[r]) * rs_s[r] * g[c] + be[c];
        C[(size_t)(tileM * 16 + r) * DOUT + c] = y > 0.0f ? y : 0.0f;
    }
}

// ---------------------------------------------------------------------------
// Segment-mean pooling over batch ids.
// ---------------------------------------------------------------------------
__global__ __launch_bounds__(128)
void pool_acc_kernel(const float* __restrict__ h,
                     const int* __restrict__ batch,
                     float* __restrict__ sums,
                     float* __restrict__ cnt,
                     int d) {
    const int row = blockIdx.x;
    const int t   = threadIdx.x;
    const int gid = batch[row];
    atomicAdd(&sums[(size_t)gid * d + t], h[(size_t)row * d + t]);
    if (t == 0) atomicAdd(&cnt[gid], 1.0f);
}

__global__ __launch_bounds__(128)
void pool_fin_kernel(const float* __restrict__ sums,
                     const float* __restrict__ cnt,
                     float* __restrict__ out,
                     int d) {
    const int gid = blockIdx.x;
    const int t   = threadIdx.x;
    float c = cnt[gid];
    c = c < 1.0f ? 1.0f : c;
    out[(size_t)gid * d + t] = sums[(size_t)gid * d + t] / c;
}

// ---------------------------------------------------------------------------
extern "C" void kernel_launch(void* const* d_in, const int* in_sizes, int n_in,
                              void* d_out, int out_size, void* d_ws, size_t ws_size,
                              hipStream_t stream) {
    const float* x      = (const float*)d_in[0];
    const int*   ei     = (const int*)d_in[1];
    const int*   batch  = (const int*)d_in[2];
    const float* Wrel1  = (const float*)d_in[3];
    const float* Wroot1 = (const float*)d_in[4];
    const float* b1     = (const float*)d_in[5];
    const float* g1     = (const float*)d_in[6];
    const float* be1    = (const float*)d_in[7];
    const float* Wrel2  = (const float*)d_in[8];
    const float* Wroot2 = (const float*)d_in[9];
    const float* b2     = (const float*)d_in[10];
    const float* g2     = (const float*)d_in[11];
    const float* be2    = (const float*)d_in[12];
    const float* Wrel3  = (const float*)d_in[13];
    const float* Wroot3 = (const float*)d_in[14];
    const float* b3     = (const float*)d_in[15];
    const float* g3     = (const float*)d_in[16];
    const float* be3    = (const float*)d_in[17];

    const int* src = ei;
    const int* dst = ei + N_EDGES;

    // Workspace layout (bytes)
    const size_t SZ_BIG = (size_t)N_NODES * 256 * sizeof(float); // 102,400,000
    char* ws = (char*)d_ws;
    float* agg  = (float*)(ws);                 // [N,256] max, reused each layer
    float* t1   = (float*)(ws + SZ_BIG);        // [N,256]; reused as t3 [N,128]
    float* t2   = (float*)(ws + 2 * SZ_BIG);    // [N,256]
    float* sums = (float*)(ws + 3 * SZ_BIG);    // [64,128]
    float* cnt  = (float*)(ws + 3 * SZ_BIG + (size_t)N_GRAPHS * 128 * sizeof(float));

    const int TILE_M = N_NODES / 16;            // 6250

    // ---------------- Layer 1: 128 -> 256 ----------------
    hipMemsetAsync(agg, 0, (size_t)N_NODES * 128 * sizeof(float), stream);
    {
        long long total = (long long)N_EDGES << 7;  // *128
        scatter_add_kernel<<<(unsigned)((total + 255) / 256), 256, 0, stream>>>(
            x, src, dst, agg, N_EDGES, 7);
    }
    gemm_ln_relu_kernel<256><<<TILE_M, 128, 0, stream>>>(
        agg, x, Wrel1, Wroot1, b1, g1, be1, t1, 128);

    // ---------------- Layer 2: 256 -> 256 ----------------
    hipMemsetAsync(agg, 0, (size_t)N_NODES * 256 * sizeof(float), stream);
    {
        long long total = (long long)N_EDGES << 8;  // *256
        scatter_add_kernel<<<(unsigned)((total + 255) / 256), 256, 0, stream>>>(
            t1, src, dst, agg, N_EDGES, 8);
    }
    gemm_ln_relu_kernel<256><<<TILE_M, 128, 0, stream>>>(
        agg, t1, Wrel2, Wroot2, b2, g2, be2, t2, 256);

    // ---------------- Layer 3: 256 -> 128 ----------------
    hipMemsetAsync(agg, 0, (size_t)N_NODES * 256 * sizeof(float), stream);
    {
        long long total = (long long)N_EDGES << 8;  // *256
        scatter_add_kernel<<<(unsigned)((total + 255) / 256), 256, 0, stream>>>(
            t2, src, dst, agg, N_EDGES, 8);
    }
    float* t3 = t1;  // reuse
    gemm_ln_relu_kernel<128><<<TILE_M, 128, 0, stream>>>(
        agg, t2, Wrel3, Wroot3, b3, g3, be3, t3, 256);

    // ---------------- Pooling: segment mean ----------------
    hipMemsetAsync(sums, 0, (size_t)N_GRAPHS * 128 * sizeof(float), stream);
    hipMemsetAsync(cnt, 0, (size_t)N_GRAPHS * sizeof(float), stream);
    pool_acc_kernel<<<N_NODES, 128, 0, stream>>>(t3, batch, sums, cnt, 128);
    pool_fin_kernel<<<N_GRAPHS, 128, 0, stream>>>(sums, cnt, (float*)d_out, 128);
}